// AlignmentContrastiveLoss_27470610825815
// MI455X (gfx1250) — compile-verified
//
#include <hip/hip_runtime.h>
#include <hip/hip_bf16.h>
#include <math.h>

// ---------------------------------------------------------------------------
// AlignmentContrastiveLoss fused kernel for MI455X (gfx1250, wave32, WMMA).
//
// S = En @ En^T (En = L2-normalized embeddings) computed tile-wise with
// v_wmma_f32_16x16x4_f32 and consumed in-register (masked pos-pair sums and
// shifted logsumexp). S (256 MB) is never materialized.
//
// Round-2 changes: K-loop in groups of 16 (4 WMMA K-steps) with compile-time
// sub-offsets -> loads use the immediate offset field (no per-step 64-bit
// address adds), 36 independent loads batched per group, then 32 WMMAs
// back-to-back with accumulator chains 8 apart.
// ---------------------------------------------------------------------------

typedef float v2f __attribute__((ext_vector_type(2)));
typedef float v8f __attribute__((ext_vector_type(8)));

#define LMAX 2048          // label range per reference setup
#define GMAX 16            // graph-id range per reference setup
#define INV_T 10.0f        // 1 / TEMPERATURE
#define LSE_SHIFT 10.0f    // max possible logit = (max cos sim)/T = 10

// ---- zero a range of ints (cnt table + accumulators, contiguous) ----------
__global__ void zero_kernel(int* __restrict__ p, int n) {
  int i = blockIdx.x * blockDim.x + threadIdx.x;
  if (i < n) p[i] = 0;
}

// ---- row L2 normalization: one wave per row -------------------------------
__global__ __launch_bounds__(256) void norm_kernel(
    const float* __restrict__ emb, float* __restrict__ E, int N, int D) {
  int row = blockIdx.x * 8 + (threadIdx.x >> 5);
  int lane = threadIdx.x & 31;
  if (row >= N) return;
  const float* src = emb + (size_t)row * D;
  float ss = 0.f;
  for (int c = lane; c < D; c += 32) { float x = src[c]; ss = fmaf(x, x, ss); }
#pragma unroll
  for (int m = 16; m; m >>= 1) ss += __shfl_xor(ss, m, 32);
  float sc = 1.f / fmaxf(sqrtf(ss), 1e-12f);
  float* dst = E + (size_t)row * D;
  for (int c = lane; c < D; c += 32) dst[c] = src[c] * sc;
}

// ---- histogram of conserved nodes per (label, graph) ----------------------
__global__ void count_kernel(const int* __restrict__ labels,
                             const int* __restrict__ graphs,
                             const int* __restrict__ cats,
                             int* __restrict__ cnt, int N) {
  int i = blockIdx.x * blockDim.x + threadIdx.x;
  if (i < N && cats[i] < 3) atomicAdd(&cnt[labels[i] * GMAX + graphs[i]], 1);
}

// ---- participation flags: node i has >=1 positive partner -----------------
__global__ void part_kernel(const int* __restrict__ labels,
                            const int* __restrict__ graphs,
                            const int* __restrict__ cats,
                            const int* __restrict__ cnt,
                            int* __restrict__ part, int N) {
  int i = blockIdx.x * blockDim.x + threadIdx.x;
  if (i >= N) return;
  int p = 0;
  if (cats[i] < 3) {
    const int* row = cnt + labels[i] * GMAX;
    int tot = 0;
#pragma unroll
    for (int g = 0; g < GMAX; ++g) tot += row[g];
    p = (tot - row[graphs[i]]) > 0;  // same label, other graph, both conserved
  }
  part[i] = p;
}

// ---- main fused WMMA pass --------------------------------------------------
// Block: 16 rows of S, 8 waves. Wave w handles 128-wide column chunks
// w, w+8, ... (8 x 16x16 WMMA tiles). K processed in groups of 16:
// 36 fragment loads (immediate offsets) then 32 WMMAs per group.
__global__ __launch_bounds__(256) void sim_kernel(
    const float* __restrict__ E, const int* __restrict__ labels,
    const int* __restrict__ graphs, const int* __restrict__ cats,
    const int* __restrict__ part, float* __restrict__ acc, int N, int D) {
  const int tid  = threadIdx.x;
  const int lane = tid & 31;
  const int wv   = tid >> 5;    // wave 0..7
  const int hi   = lane >> 4;   // lane half: selects K pair (A/B) / row+8 (C)
  const int l16  = lane & 15;
  const int i0   = blockIdx.x * 16;

  __shared__ float row_s[16], row_ps[16], row_pc[16];
  if (tid < 16) { row_s[tid] = 0.f; row_ps[tid] = 0.f; row_pc[tid] = 0.f; }
  __syncthreads();

  // attributes of this lane's 8 rows (C VGPR v -> row i0 + v + 8*hi)
  int lr[8], gr[8]; bool cr[8], pr[8];
#pragma unroll
  for (int v = 0; v < 8; ++v) {
    int r = i0 + v + 8 * hi;
    lr[v] = labels[r]; gr[v] = graphs[r];
    cr[v] = cats[r] < 3; pr[v] = part[r] != 0;
  }

  float s_acc[8], ps_acc[8], pc_acc[8];
#pragma unroll
  for (int v = 0; v < 8; ++v) { s_acc[v] = 0.f; ps_acc[v] = 0.f; pc_acc[v] = 0.f; }

  // A fragment base: lane holds row i0+l16, K pair {k+2*hi, k+2*hi+1}
  const float* arow = E + (size_t)(i0 + l16) * D + 2 * hi;

  const int nchunk = N >> 7;  // 128-column chunks
  for (int ch = wv; ch < nchunk; ch += 8) {
    const int j0 = ch << 7;

    // row pointers for the 8 sub-tiles; bumped once per 16-K group
    const float* ap = arow;
    const float* bp[8];
#pragma unroll
    for (int t = 0; t < 8; ++t)
      bp[t] = E + (size_t)(j0 + 16 * t + l16) * D + 2 * hi;

    v8f cfr[8] = {};

    for (int kg = 0; kg < D; kg += 16) {
      // ---- stage all 36 fragments for this group (immediate offsets) ----
      v2f a0 = *(const v2f*)(ap + 0);
      v2f a1 = *(const v2f*)(ap + 4);
      v2f a2 = *(const v2f*)(ap + 8);
      v2f a3 = *(const v2f*)(ap + 12);
      v2f b0[8], b1[8], b2[8], b3[8];
#pragma unroll
      for (int t = 0; t < 8; ++t) {
        b0[t] = *(const v2f*)(bp[t] + 0);
        b1[t] = *(const v2f*)(bp[t] + 4);
        b2[t] = *(const v2f*)(bp[t] + 8);
        b3[t] = *(const v2f*)(bp[t] + 12);
      }
      // ---- 32 matrix ops; dependent chains (same cfr[t]) are 8 apart ----
#pragma unroll
      for (int t = 0; t < 8; ++t)
        cfr[t] = __builtin_amdgcn_wmma_f32_16x16x4_f32(
            false, a0, false, b0[t], (short)0, cfr[t], false, false);
#pragma unroll
      for (int t = 0; t < 8; ++t)
        cfr[t] = __builtin_amdgcn_wmma_f32_16x16x4_f32(
            false, a1, false, b1[t], (short)0, cfr[t], false, false);
#pragma unroll
      for (int t = 0; t < 8; ++t)
        cfr[t] = __builtin_amdgcn_wmma_f32_16x16x4_f32(
            false, a2, false, b2[t], (short)0, cfr[t], false, false);
#pragma unroll
      for (int t = 0; t < 8; ++t)
        cfr[t] = __builtin_amdgcn_wmma_f32_16x16x4_f32(
            false, a3, false, b3[t], (short)0, cfr[t], false, false);

      ap += 16;
#pragma unroll
      for (int t = 0; t < 8; ++t) bp[t] += 16;
    }

    // epilogue: masks + shifted sum-exp + positive sums, all in-register
#pragma unroll
    for (int t = 0; t < 8; ++t) {
      const int  c    = j0 + 16 * t + l16;  // this lane's column
      const int  lc   = labels[c], gc = graphs[c];
      const bool cc   = cats[c] < 3;
      const bool pcol = part[c] != 0;
#pragma unroll
      for (int v = 0; v < 8; ++v) {
        const float sv = cfr[t][v];
        const int   r  = i0 + v + 8 * hi;
        const bool sg = (gr[v] == gc), sl = (lr[v] == lc);
        const bool pos = sl && !sg && cr[v] && cc;
        const bool neg = !sg && !sl;
        const bool valid = (pos || neg) && pr[v] && pcol && (r != c);
        if (valid) s_acc[v] += __expf(fmaf(INV_T, sv, -LSE_SHIFT));
        if (pos)  { pc_acc[v] += 1.f; ps_acc[v] += sv; }
      }
    }
  }

  // reduce each row across the 16 lanes that share it, then merge waves in LDS
#pragma unroll
  for (int v = 0; v < 8; ++v) {
    float x = s_acc[v], y = ps_acc[v], z = pc_acc[v];
#pragma unroll
    for (int m = 8; m; m >>= 1) {
      x += __shfl_xor(x, m, 16);
      y += __shfl_xor(y, m, 16);
      z += __shfl_xor(z, m, 16);
    }
    if (l16 == 0) {
      atomicAdd(&row_s[v + 8 * hi], x);
      atomicAdd(&row_ps[v + 8 * hi], y);
      atomicAdd(&row_pc[v + 8 * hi], z);
    }
  }
  __syncthreads();

  if (wv == 0) {
    float pc = 0.f, ps = 0.f, s = 1.f;
    if (lane < 16) { pc = row_pc[lane]; ps = row_ps[lane]; s = row_s[lane]; }
    float npos_c = pc;
    float pos_c  = pc - ps;  // sum of (1 - S) over positives in these rows
    // sum over positives of (lse_i - S_ij/T) = pc*lse - ps/T
    float nce_c = (pc > 0.f)
                      ? fmaf(pc, LSE_SHIFT + __logf(s), -INV_T * ps)
                      : 0.f;
#pragma unroll
    for (int m = 16; m; m >>= 1) {
      npos_c += __shfl_xor(npos_c, m, 32);
      pos_c  += __shfl_xor(pos_c, m, 32);
      nce_c  += __shfl_xor(nce_c, m, 32);
    }
    if (lane == 0) {
      atomicAdd(&acc[0], npos_c);
      atomicAdd(&acc[1], pos_c);
      atomicAdd(&acc[2], nce_c);
    }
  }
}

// ---- finalize: pos_loss + nce_loss ----------------------------------------
// n_pairs = n_pos/2 and upper-tri sum = full sum / 2 (pos mask is symmetric,
// diag-free), so pos_loss = acc[1]/acc[0]; nce_loss = acc[2]/acc[0].
__global__ void final_kernel(const float* __restrict__ acc, float* __restrict__ out) {
  float np = acc[0];
  out[0] = (np > 0.f) ? (acc[1] + acc[2]) / np : 0.f;
}

extern "C" void kernel_launch(void* const* d_in, const int* in_sizes, int n_in,
                              void* d_out, int out_size, void* d_ws, size_t ws_size,
                              hipStream_t stream) {
  (void)n_in; (void)out_size; (void)ws_size;
  const float* emb    = (const float*)d_in[0];
  const int*   labels = (const int*)d_in[1];
  const int*   graphs = (const int*)d_in[2];
  const int*   cats   = (const int*)d_in[3];
  const int N = in_sizes[1];
  const int D = in_sizes[0] / N;  // 8192 x 256

  // workspace layout: [ E : N*D f32 ][ cnt : LMAX*GMAX i32 ][ acc : 4 f32 ][ part : N i32 ]
  char*  ws  = (char*)d_ws;
  float* E   = (float*)ws;
  size_t eb  = (size_t)N * D * sizeof(float);
  int*   cnt = (int*)(ws + eb);
  float* acc = (float*)(ws + eb + (size_t)LMAX * GMAX * sizeof(int));
  int*   part= (int*)(ws + eb + (size_t)LMAX * GMAX * sizeof(int) + 4 * sizeof(float));

  const int zn = LMAX * GMAX + 4;  // cnt table + acc (contiguous)
  zero_kernel <<<(zn + 255) / 256, 256, 0, stream>>>(cnt, zn);
  norm_kernel <<<(N + 7) / 8,      256, 0, stream>>>(emb, E, N, D);
  count_kernel<<<(N + 255) / 256,  256, 0, stream>>>(labels, graphs, cats, cnt, N);
  part_kernel <<<(N + 255) / 256,  256, 0, stream>>>(labels, graphs, cats, cnt, part, N);
  sim_kernel  <<<N / 16,           256, 0, stream>>>(E, labels, graphs, cats, part, acc, N, D);
  final_kernel<<<1, 1, 0, stream>>>(acc, (float*)d_out);
}